// Regressor_GNN_23682449670460
// MI455X (gfx1250) — compile-verified
//
#include <hip/hip_runtime.h>
#include <hip/hip_bf16.h>
#include <math.h>

#define BN_EPS 1e-5f
#define NEG_SLOPE 0.2f

typedef float v2f __attribute__((ext_vector_type(2)));
typedef float v8f __attribute__((ext_vector_type(8)));

// ---------------- utility kernels ----------------
__global__ void fill_f32(float* p, float v, int n) {
  int i = blockIdx.x * blockDim.x + threadIdx.x;
  if (i < n) p[i] = v;
}
__global__ void fill_u32(unsigned* p, unsigned v, int n) {
  int i = blockIdx.x * blockDim.x + threadIdx.x;
  if (i < n) p[i] = v;
}

// order-preserving float<->uint map so atomicMax(u32) == float max (handles negatives)
__device__ __forceinline__ unsigned mapf(float f) {
  unsigned u = __float_as_uint(f);
  return (u & 0x80000000u) ? ~u : (u | 0x80000000u);
}
__device__ __forceinline__ float unmapf(unsigned u) {
  u = (u & 0x80000000u) ? (u & 0x7fffffffu) : ~u;
  return __uint_as_float(u);
}

// ---------------- batchnorm ----------------
__global__ void bn_stats(const float* __restrict__ x, float* __restrict__ mean,
                         float* __restrict__ var, int Nn, int D) {
  int j = blockIdx.x;
  if (j >= D) return;
  __shared__ float s1[256];
  __shared__ float s2[256];
  float a = 0.f, b = 0.f;
  for (int i = threadIdx.x; i < Nn; i += blockDim.x) {
    float v = x[(size_t)i * D + j];
    a += v; b += v * v;
  }
  s1[threadIdx.x] = a; s2[threadIdx.x] = b;
  __syncthreads();
  for (int s = 128; s > 0; s >>= 1) {
    if ((int)threadIdx.x < s) {
      s1[threadIdx.x] += s1[threadIdx.x + s];
      s2[threadIdx.x] += s2[threadIdx.x + s];
    }
    __syncthreads();
  }
  if (threadIdx.x == 0) {
    float mu = s1[0] / (float)Nn;
    mean[j] = mu;
    var[j] = s2[0] / (float)Nn - mu * mu;
  }
}

__global__ void bn_apply(const float* __restrict__ x, const float* __restrict__ mean,
                         const float* __restrict__ var, const float* __restrict__ g,
                         const float* __restrict__ b, float* __restrict__ out,
                         int total, int D, int relu) {
  int idx = blockIdx.x * blockDim.x + threadIdx.x;
  if (idx >= total) return;
  int j = idx % D;
  float v = (x[idx] - mean[j]) * rsqrtf(var[j] + BN_EPS) * g[j] + b[j];
  if (relu && v < 0.f) v = 0.f;
  out[idx] = v;
}

// ---------------- WMMA f32 GEMM: C[MxN] = A[MxK] @ B[KxN], row-major ----------------
// One wave32 computes one 16x16 output tile via V_WMMA_F32_16X16X4_F32.
// A-lane layout (ISA 7.12.2): lane l row = l%16; VGPR0/1 hold K = k0+{0,1} (lanes 0-15)
// or k0+{2,3} (lanes 16-31). B mirrors with N = lane%16. C/D: row = r + 8*(lane>=16).
// Steady-state K loop is unpredicated (tail peeled) so the inner loop is pure
// global_load_b32 + v_wmma with no exec juggling.
__global__ void gemm_wmma_f32(const float* __restrict__ A, const float* __restrict__ B,
                              float* __restrict__ C, int M, int K, int N) {
  int gtid = blockIdx.x * blockDim.x + threadIdx.x;
  int wave = gtid >> 5;
  int lane = threadIdx.x & 31;
  int ntiles = N >> 4;                 // N is a multiple of 16
  int mtiles = (M + 15) >> 4;
  if (wave >= mtiles * ntiles) return; // wave-uniform exit: EXEC all-ones below
  int mt = wave / ntiles;
  int nt = wave - mt * ntiles;
  int l15 = lane & 15;
  int m = mt * 16 + l15;               // A row for this lane
  int n = nt * 16 + l15;               // B col for this lane
  int kk = (lane < 16) ? 0 : 2;

  // clamp out-of-range rows: a clamped row only affects its own (never stored) D row
  int mr = (m < M) ? m : (M - 1);
  const float* Arow = A + (size_t)mr * K + kk;
  const float* Bcol = B + (size_t)kk * N + n;

  int Kmain = K & ~3;
  v8f acc = {};
#pragma unroll 4
  for (int k0 = 0; k0 < Kmain; k0 += 4) {
    v2f a, b;
    a.x = Arow[k0];
    a.y = Arow[k0 + 1];
    b.x = Bcol[(size_t)k0 * N];
    b.y = Bcol[(size_t)(k0 + 1) * N];
    acc = __builtin_amdgcn_wmma_f32_16x16x4_f32(
        false, a, false, b, (short)0, acc, false, false);
  }
  if (Kmain < K) {  // single predicated tail step (K % 4 != 0, e.g. K=129)
    int ka = Kmain + kk;
    v2f a, b;
    a.x = (ka < K)       ? Arow[Kmain]                      : 0.f;
    a.y = ((ka + 1) < K) ? Arow[Kmain + 1]                  : 0.f;
    b.x = (ka < K)       ? Bcol[(size_t)Kmain * N]          : 0.f;
    b.y = ((ka + 1) < K) ? Bcol[(size_t)(Kmain + 1) * N]    : 0.f;
    acc = __builtin_amdgcn_wmma_f32_16x16x4_f32(
        false, a, false, b, (short)0, acc, false, false);
  }

  int rbase = mt * 16 + ((lane >= 16) ? 8 : 0);
#pragma unroll
  for (int r = 0; r < 8; ++r) {
    int row = rbase + r;
    if (row < M) C[(size_t)row * N + n] = acc[r];
  }
}

// N x 1 projection for layer 3 (GEMM with 1 output column -> plain dot)
__global__ void matvec(const float* __restrict__ A, const float* __restrict__ w,
                       float* __restrict__ out, int Nn, int K) {
  int i = blockIdx.x * blockDim.x + threadIdx.x;
  if (i >= Nn) return;
  const float* row = A + (size_t)i * K;
  float acc = 0.f;
  for (int k = 0; k < K; ++k) acc += row[k] * w[k];
  out[i] = acc;
}

// ---------------- GATv2 edge phase ----------------
__global__ void edge_logits(const float* __restrict__ xl, const float* __restrict__ xr,
                            const int* __restrict__ src, const int* __restrict__ dst,
                            const float* __restrict__ att,
                            float* __restrict__ logits, unsigned* __restrict__ mmax,
                            int E, int H, int C) {
  int idx = blockIdx.x * blockDim.x + threadIdx.x;
  if (idx >= E * H) return;
  int e = idx / H, h = idx - e * H;
  int sN = src[e], dN = dst[e];
  const float* pl = xl + (size_t)sN * H * C + (size_t)h * C;
  const float* pr = xr + (size_t)dN * H * C + (size_t)h * C;
  const float* pa = att + (size_t)h * C;
  float acc = 0.f;
  if ((C & 3) == 0) {  // rows are 16B-aligned (arena 256B-aligned, C multiple of 4)
    const float4* pl4 = (const float4*)pl;
    const float4* pr4 = (const float4*)pr;
    const float4* pa4 = (const float4*)pa;
    int C4 = C >> 2;
    for (int c = 0; c < C4; ++c) {
      float4 l = pl4[c], r = pr4[c], aa = pa4[c];
      float v0 = l.x + r.x; v0 = (v0 > 0.f) ? v0 : v0 * NEG_SLOPE;
      float v1 = l.y + r.y; v1 = (v1 > 0.f) ? v1 : v1 * NEG_SLOPE;
      float v2 = l.z + r.z; v2 = (v2 > 0.f) ? v2 : v2 * NEG_SLOPE;
      float v3 = l.w + r.w; v3 = (v3 > 0.f) ? v3 : v3 * NEG_SLOPE;
      acc += aa.x * v0 + aa.y * v1 + aa.z * v2 + aa.w * v3;
    }
  } else {
    for (int c = 0; c < C; ++c) {
      float v = pl[c] + pr[c];
      v = (v > 0.f) ? v : v * NEG_SLOPE;
      acc += pa[c] * v;
    }
  }
  logits[idx] = acc;
  atomicMax(&mmax[(size_t)dN * H + h], mapf(acc));
}

// in-place: logits -> exp(logit - segmax); accumulate denominator and degree
__global__ void edge_softmax(float* __restrict__ logits, const int* __restrict__ dst,
                             const unsigned* __restrict__ mmax, float* __restrict__ ssum,
                             float* __restrict__ deg, int E, int H) {
  int idx = blockIdx.x * blockDim.x + threadIdx.x;
  if (idx >= E * H) return;
  int e = idx / H, h = idx - e * H;
  int dN = dst[e];
  float m = unmapf(mmax[(size_t)dN * H + h]);
  float a = expf(logits[idx] - m);
  logits[idx] = a;
  atomicAdd(&ssum[(size_t)dN * H + h], a);
  if (h == 0) atomicAdd(&deg[dN], 1.0f);
}

// summ[dst] += xl[src] * alpha  (one thread per edge-feature)
__global__ void edge_scatter(const float* __restrict__ xl, const int* __restrict__ src,
                             const int* __restrict__ dst, const float* __restrict__ aexp,
                             const float* __restrict__ ssum, float* __restrict__ summ,
                             int E, int H, int C) {
  int idx = blockIdx.x * blockDim.x + threadIdx.x;
  int HC = H * C;
  if (idx >= E * HC) return;
  int e = idx / HC;
  int r = idx - e * HC;
  int h = r / C;
  int sN = src[e], dN = dst[e];
  float alpha = aexp[(size_t)e * H + h] / ssum[(size_t)dN * H + h];
  atomicAdd(&summ[(size_t)dN * HC + r], xl[(size_t)sN * HC + r] * alpha);
}

__global__ void node_finalize(const float* __restrict__ summ, const float* __restrict__ deg,
                              const float* __restrict__ bias, float* __restrict__ out,
                              int Nn, int HC) {
  int idx = blockIdx.x * blockDim.x + threadIdx.x;
  if (idx >= Nn * HC) return;
  int n = idx / HC, j = idx - n * HC;
  float dg = deg[n];
  if (dg < 1.f) dg = 1.f;
  out[idx] = summ[idx] / dg + bias[j];
}

__global__ void gather_out(const float* __restrict__ node_out, const float* __restrict__ y,
                           const float* __restrict__ w, const int* __restrict__ tidx,
                           float* __restrict__ dout, int NT) {
  int i = blockIdx.x * blockDim.x + threadIdx.x;
  if (i >= NT) return;
  int n = tidx[i];
  dout[i] = node_out[n];
  dout[NT + i] = y[n];
  dout[2 * NT + i] = w[n];
}

// ---------------- host orchestration ----------------
extern "C" void kernel_launch(void* const* d_in, const int* in_sizes, int n_in,
                              void* d_out, int out_size, void* d_ws, size_t ws_size,
                              hipStream_t stream) {
  (void)n_in; (void)out_size; (void)ws_size;
  const float* x      = (const float*)d_in[0];
  const int* eidx     = (const int*)d_in[1];
  const float* y      = (const float*)d_in[2];
  const float* w      = (const float*)d_in[3];
  const int* tidx     = (const int*)d_in[4];
  const float* bn1_g  = (const float*)d_in[5];
  const float* bn1_b  = (const float*)d_in[6];
  const float* W1l    = (const float*)d_in[7];
  const float* W1r    = (const float*)d_in[8];
  const float* att1   = (const float*)d_in[9];
  const float* b1     = (const float*)d_in[10];
  const float* bn2_g  = (const float*)d_in[11];
  const float* bn2_b  = (const float*)d_in[12];
  const float* W2l    = (const float*)d_in[13];
  const float* W2r    = (const float*)d_in[14];
  const float* att2   = (const float*)d_in[15];
  const float* b2     = (const float*)d_in[16];
  const float* bn3_g  = (const float*)d_in[17];
  const float* bn3_b  = (const float*)d_in[18];
  const float* W3l    = (const float*)d_in[19];
  const float* W3r    = (const float*)d_in[20];
  const float* att3   = (const float*)d_in[21];
  const float* b3     = (const float*)d_in[22];

  const int Nn = in_sizes[2];           // 50000
  const int D  = in_sizes[0] / Nn;      // 129
  const int E  = in_sizes[1] / 2;       // 800000
  const int NT = in_sizes[4];           // 10000
  const int* src = eidx;
  const int* dst = eidx + E;

  // workspace arena (256B aligned slots)
  char* base = (char*)d_ws;
  size_t off = 0;
  auto arena = [&](size_t bytes) -> char* {
    size_t o = off;
    off += (bytes + 255) & ~(size_t)255;
    return base + o;
  };
  float*    meanb = (float*)arena(256 * sizeof(float));
  float*    varb  = (float*)arena(256 * sizeof(float));
  float*    hbuf  = (float*)arena((size_t)Nn * 256 * sizeof(float));
  float*    outb  = (float*)arena((size_t)Nn * 256 * sizeof(float));
  float*    xlb   = (float*)arena((size_t)Nn * 256 * sizeof(float));
  float*    xrb   = (float*)arena((size_t)Nn * 256 * sizeof(float));
  float*    logb  = (float*)arena((size_t)E * 2 * sizeof(float));
  unsigned* mmaxb = (unsigned*)arena((size_t)Nn * 2 * sizeof(unsigned));
  float*    ssumb = (float*)arena((size_t)Nn * 2 * sizeof(float));
  float*    degb  = (float*)arena((size_t)Nn * sizeof(float));
  float*    summb = xrb;  // xr is dead after edge_logits -> reuse as scatter accumulator

  const int TPB = 256;
  auto nb = [](long long n) { return (unsigned)((n + 255) / 256); };

  auto gat_edges = [&](const float* att, const float* bias, int H, int C) {
    fill_u32<<<nb(Nn * H), TPB, 0, stream>>>(mmaxb, 0u, Nn * H);
    fill_f32<<<nb(Nn * H), TPB, 0, stream>>>(ssumb, 0.f, Nn * H);
    fill_f32<<<nb(Nn), TPB, 0, stream>>>(degb, 0.f, Nn);
    edge_logits<<<nb((long long)E * H), TPB, 0, stream>>>(xlb, xrb, src, dst, att, logb, mmaxb, E, H, C);
    edge_softmax<<<nb((long long)E * H), TPB, 0, stream>>>(logb, dst, mmaxb, ssumb, degb, E, H);
    fill_f32<<<nb((long long)Nn * H * C), TPB, 0, stream>>>(summb, 0.f, Nn * H * C);
    edge_scatter<<<nb((long long)E * H * C), TPB, 0, stream>>>(xlb, src, dst, logb, ssumb, summb, E, H, C);
    node_finalize<<<nb((long long)Nn * H * C), TPB, 0, stream>>>(summb, degb, bias, outb, Nn, H * C);
  };
  auto gemm = [&](const float* A, const float* B, float* C, int M, int K, int Ncol) {
    long long waves = (long long)((M + 15) / 16) * (Ncol / 16);
    gemm_wmma_f32<<<nb(waves * 32), TPB, 0, stream>>>(A, B, C, M, K, Ncol);
  };

  // ---- layer 1: BN -> GATv2(129 -> 2 heads x 128) ----
  bn_stats<<<D, TPB, 0, stream>>>(x, meanb, varb, Nn, D);
  bn_apply<<<nb((long long)Nn * D), TPB, 0, stream>>>(x, meanb, varb, bn1_g, bn1_b, hbuf, Nn * D, D, 0);
  gemm(hbuf, W1l, xlb, Nn, D, 256);
  gemm(hbuf, W1r, xrb, Nn, D, 256);
  gat_edges(att1, b1, 2, 128);

  // ---- layer 2: ReLU(BN) -> GATv2(256 -> 1 head x 128) ----
  bn_stats<<<256, TPB, 0, stream>>>(outb, meanb, varb, Nn, 256);
  bn_apply<<<nb((long long)Nn * 256), TPB, 0, stream>>>(outb, meanb, varb, bn2_g, bn2_b, hbuf, Nn * 256, 256, 1);
  gemm(hbuf, W2l, xlb, Nn, 256, 128);
  gemm(hbuf, W2r, xrb, Nn, 256, 128);
  gat_edges(att2, b2, 1, 128);

  // ---- layer 3: ReLU(BN) -> GATv2(128 -> 1 head x 1) ----
  bn_stats<<<128, TPB, 0, stream>>>(outb, meanb, varb, Nn, 128);
  bn_apply<<<nb((long long)Nn * 128), TPB, 0, stream>>>(outb, meanb, varb, bn3_g, bn3_b, hbuf, Nn * 128, 128, 1);
  matvec<<<nb(Nn), TPB, 0, stream>>>(hbuf, W3l, xlb, Nn, 128);
  matvec<<<nb(Nn), TPB, 0, stream>>>(hbuf, W3r, xrb, Nn, 128);
  gat_edges(att3, b3, 1, 1);

  // ---- gather train_idx outputs: [pred | y | w] ----
  gather_out<<<nb(NT), TPB, 0, stream>>>(outb, y, w, tidx, (float*)d_out, NT);
}